// TextVAE_69801808494710
// MI455X (gfx1250) — compile-verified
//
#include <hip/hip_runtime.h>
#include <cmath>

// ---------------------------------------------------------------------------
// TextVAE loss for MI455X (gfx1250, wave32, WMMA).
//   - f32 WMMA 16x16x4 for all GEMMs (keeps reference f32 precision).
//   - Fused vocab GEMMs + online logsumexp (never materialize 4096x10000).
//   - 64 rows/WG, 2 M-tiles per wave sharing one B fragment (load reuse).
//   - TDM tensor_load_to_lds double-buffered prefetch of per-tile bias rows.
//   - Persistent LSTM kernel (32 WGs) with global barrier between time steps.
// ---------------------------------------------------------------------------

typedef float v2f __attribute__((ext_vector_type(2)));
typedef float v8f __attribute__((ext_vector_type(8)));
typedef unsigned int v4u __attribute__((ext_vector_type(4)));
typedef int v4i __attribute__((ext_vector_type(4)));
typedef int v8i __attribute__((ext_vector_type(8)));

#define DEV static __device__ __forceinline__

#if defined(__has_builtin)
#if __has_builtin(__builtin_amdgcn_tensor_load_to_lds) && \
    __has_builtin(__builtin_amdgcn_s_wait_tensorcnt)
#define USE_TDM 1
#endif
#endif

constexpr int cB  = 32;
constexpr int cT  = 128;
constexpr int cH  = 512;
constexpr int cG  = 2048;   // 4*H
constexpr int cBT = 4096;   // B*T
constexpr int cV  = 10000;
constexpr float kHalfLog2Pi = 0.91893853320467274f;

DEV float softplus_b(float x) {
  // softplus(log2 * x) / log2
  float a = 0.69314718055994531f * x;
  float sp = (a > 20.f) ? a : log1pf(expf(a));
  return sp * 1.4426950408889634f;
}

DEV float sigm(float x) { return 1.f / (1.f + expf(-x)); }

// One 16x16 f32 C-tile.  ap/bp include the lane's m/n row and koff offsets per
// the CDNA5 WMMA f32 16x16x4 layout:
//   lane l: m = n = l&15 ; koff = (l>>4)*2 ; frag = {X[r][kb+koff], X[r][kb+koff+1]}
DEV v8f wmma_span(const float* __restrict__ ap, const float* __restrict__ bp,
                  int Kdim, v8f acc) {
#pragma unroll 8
  for (int kb = 0; kb < Kdim; kb += 4) {
    v2f a = *(const v2f*)(ap + kb);
    v2f b = *(const v2f*)(bp + kb);
    acc = __builtin_amdgcn_wmma_f32_16x16x4_f32(false, a, false, b,
                                                (short)0, acc, false, false);
  }
  return acc;
}

// Two M-tiles sharing one streamed B fragment (cuts fragment loads 25%).
DEV void wmma_span2(const float* __restrict__ a0p, const float* __restrict__ a1p,
                    const float* __restrict__ bp, int Kdim,
                    v8f& acc0, v8f& acc1) {
#pragma unroll 4
  for (int kb = 0; kb < Kdim; kb += 4) {
    v2f b  = *(const v2f*)(bp + kb);
    v2f a0 = *(const v2f*)(a0p + kb);
    v2f a1 = *(const v2f*)(a1p + kb);
    acc0 = __builtin_amdgcn_wmma_f32_16x16x4_f32(false, a0, false, b,
                                                 (short)0, acc0, false, false);
    acc1 = __builtin_amdgcn_wmma_f32_16x16x4_f32(false, a1, false, b,
                                                 (short)0, acc1, false, false);
  }
}

#ifdef USE_TDM
// 1-row TDM transfer: tile 64 f32, tensor_dim0 = rem (OOB zero-filled).
// D# bitfields per CDNA5 ISA (count=1, type=2=image,
// data_size=4B, 2D descriptor with dim1 = 1).  6-arg builtin form
// (clang-23 / therock headers): (v4u, v8i, v4i, v4i, v8i, i32 cpol).
DEV void tdm_load_row(const float* src, int rem, void* lds_dst) {
  unsigned lds_off = (unsigned)(uintptr_t)lds_dst;      // flat low 32 = LDS offset
  unsigned long long ga = (unsigned long long)(uintptr_t)src;
  v4u g0 = { 1u,                                        // count=1, user mode
             lds_off,                                   // lds_addr
             (unsigned)(ga & 0xFFFFFFFFull),            // global_addr[31:0]
             (unsigned)((ga >> 32) & 0x01FFFFFFull) | 0x80000000u }; // [56:32]|type=2
  v8i g1 = { (int)0x00020000u,                          // data_size=2 (4 bytes)
             (int)(((unsigned)rem & 0xFFFFu) << 16),    // tensor_dim0 low16
             (int)0x00010000u,                          // tensor_dim1 = 1
             (int)(64u << 16),                          // tile_dim0 = 64
             1,                                         // tile_dim1 = 1
             64,                                        // tensor_dim0_stride
             0, 0 };
  v4i gz4 = { 0, 0, 0, 0 };
  v8i gz8 = { 0, 0, 0, 0, 0, 0, 0, 0 };
  __builtin_amdgcn_tensor_load_to_lds(g0, g1, gz4, gz4, gz8, 0);
}
#endif

// ---------------------------------------------------------------------------
// K0: per-vocab prior coefficients.
//  Cmat[v][0:256]   = mu_p/sig_p^2      (coef of z)
//  Cmat[v][256:512] = -0.5/sig_p^2      (coef of z^2) -> lp = const + [z|z2]·Cmat[v]
//  constv[v] = sum_k(-HALF_LOG_2PI - log(sig_p) - mu_p^2/(2 sig_p^2))
// ---------------------------------------------------------------------------
__global__ void __launch_bounds__(256) prior_kernel(const float* __restrict__ emb,
                                                    float* __restrict__ Cmat,
                                                    float* __restrict__ constv) {
  int v = blockIdx.x, k = threadIdx.x;
  float mu  = emb[(size_t)v * 512 + k];
  float sig = softplus_b(emb[(size_t)v * 512 + 256 + k]);
  float iv  = 0.5f / (sig * sig);
  Cmat[(size_t)v * 512 + k]       = 2.f * iv * mu;
  Cmat[(size_t)v * 512 + 256 + k] = -iv;
  float part = -kHalfLog2Pi - logf(sig) - mu * mu * iv;

  __shared__ float red[256];
  red[k] = part;
  __syncthreads();
  for (int s = 128; s > 0; s >>= 1) {
    if (k < s) red[k] += red[k + s];
    __syncthreads();
  }
  if (k == 0) constv[v] = red[0];
}

// ---------------------------------------------------------------------------
// K1: z / z^2 / q_z_lp.   zc[row][0:256]=z, zc[row][256:512]=z*z
// ---------------------------------------------------------------------------
__global__ void __launch_bounds__(256) z_kernel(const int* __restrict__ xg,
                                                const int* __restrict__ xsl,
                                                const float* __restrict__ eps,
                                                const float* __restrict__ emb,
                                                float* __restrict__ zc,
                                                float* __restrict__ qz) {
  int r = blockIdx.x, k = threadIdx.x;
  int b = r >> 7, tt = r & 127;
  int xv = xg[r];
  float mu  = emb[(size_t)xv * 512 + k];
  float sig = softplus_b(emb[(size_t)xv * 512 + 256 + k]);
  float ep  = eps[(size_t)r * 256 + k];
  float mask = (tt < xsl[b]) ? 1.f : 0.f;
  float zv = (mu + sig * ep) * mask;
  zc[(size_t)r * 512 + k]       = zv;
  zc[(size_t)r * 512 + 256 + k] = zv * zv;
  float qpart = -kHalfLog2Pi - logf(sig) - 0.5f * ep * ep;

  __shared__ float red[256];
  red[k] = qpart;
  __syncthreads();
  for (int s = 128; s > 0; s >>= 1) {
    if (k < s) red[k] += red[k + s];
    __syncthreads();
  }
  if (k == 0) qz[r] = red[0];
}

// ---------------------------------------------------------------------------
// K_init: zero ping-pong h buffers + barrier counter (each launch -> determinism)
// ---------------------------------------------------------------------------
__global__ void __launch_bounds__(256) init_kernel(float* hb0, float* hb1,
                                                   unsigned* ctr) {
  int id = blockIdx.x * 256 + threadIdx.x;
  if (id < cB * cH) { hb0[id] = 0.f; hb1[id] = 0.f; }
  if (id == 0) *ctr = 0u;
}

// ---------------------------------------------------------------------------
// K2: x_proj = z_shift @ w_ih.T + b_ih + b_hh   (4096 x 2048, K=256), WMMA.
// ---------------------------------------------------------------------------
__global__ void __launch_bounds__(256) xproj_kernel(const float* __restrict__ zc,
                                                    const float* __restrict__ w_ih,
                                                    const float* __restrict__ b_ih,
                                                    const float* __restrict__ b_hh,
                                                    float* __restrict__ xproj) {
  const int tid = threadIdx.x, lane = tid & 31, w = tid >> 5;
  const int mi = w & 1, ni = w >> 1;
  const int nm = lane & 15, koff = (lane >> 4) << 1, mb = (lane >> 4) << 3;
  const int row0 = blockIdx.y * 32 + mi * 16;
  const int n0   = blockIdx.x * 64 + ni * 16;

  int r  = row0 + nm;
  int tt = r & 127;
  int rsrc = tt ? (r - 1) : r;
  float scale = tt ? 1.f : 0.f;
  const float* ap = zc + (size_t)rsrc * 512 + koff;          // z half of zc
  const float* bp = w_ih + (size_t)(n0 + nm) * 256 + koff;

  v8f acc = {};
#pragma unroll 8
  for (int kb = 0; kb < 256; kb += 4) {
    v2f a = *(const v2f*)(ap + kb);
    a *= scale;
    v2f b = *(const v2f*)(bp + kb);
    acc = __builtin_amdgcn_wmma_f32_16x16x4_f32(false, a, false, b,
                                                (short)0, acc, false, false);
  }
  int g = n0 + nm;
  float bias = b_ih[g] + b_hh[g];
#pragma unroll
  for (int rr = 0; rr < 8; ++rr)
    xproj[(size_t)(row0 + mb + rr) * cG + g] = acc[rr] + bias;
}

// ---------------------------------------------------------------------------
// K3: persistent LSTM. 32 workgroups; WG wg owns h-indices [wg*16, wg*16+16).
// Per step: gates tile (M=32 batches, N=64 = 4 gates x 16 h-idx, K=512) via
// WMMA, then cell/h update, ping-pong h through global, global barrier.
// ---------------------------------------------------------------------------
__global__ void __launch_bounds__(256) lstm_kernel(const float* __restrict__ xproj,
                                                   const float* __restrict__ w_hh,
                                                   float* __restrict__ hb0,
                                                   float* __restrict__ hb1,
                                                   float* __restrict__ hs,
                                                   unsigned* __restrict__ ctr) {
  const int tid = threadIdx.x, lane = tid & 31, w = tid >> 5;
  const int mi = w & 1, gate = w >> 1;            // 8 waves: 2 batch-blocks x 4 gates
  const int nm = lane & 15, koff = (lane >> 4) << 1, mb = (lane >> 4) << 3;
  const int j0 = blockIdx.x * 16;

  __shared__ float gbuf[32][68];                  // gates tile, padded

  const int b0 = tid >> 4,          jj0 = tid & 15;
  const int b1 = (tid + 256) >> 4,  jj1 = tid & 15;
  float c0 = 0.f, c1 = 0.f;

  const float* bp = w_hh + (size_t)(gate * cH + j0 + nm) * cH + koff;

  for (int step = 0; step < cT; ++step) {
    const float* hc = (step & 1) ? hb1 : hb0;     // read buffer
    float*       hn = (step & 1) ? hb0 : hb1;     // write buffer

    const float* ap = hc + (size_t)(mi * 16 + nm) * cH + koff;
    v8f acc = {};
    acc = wmma_span(ap, bp, cH, acc);
#pragma unroll
    for (int rr = 0; rr < 8; ++rr)
      gbuf[mi * 16 + mb + rr][gate * 16 + nm] = acc[rr];
    __syncthreads();

    {
      int row = b0 * cT + step;
      const float* xp = xproj + (size_t)row * cG + j0;
      float gi = xp[0 * cH + jj0] + gbuf[b0][0 * 16 + jj0];
      float gf = xp[1 * cH + jj0] + gbuf[b0][1 * 16 + jj0];
      float gg = xp[2 * cH + jj0] + gbuf[b0][2 * 16 + jj0];
      float go = xp[3 * cH + jj0] + gbuf[b0][3 * 16 + jj0];
      c0 = sigm(gf) * c0 + sigm(gi) * tanhf(gg);
      float h = sigm(go) * tanhf(c0);
      hn[b0 * cH + j0 + jj0] = h;
      hs[(size_t)row * cH + j0 + jj0] = h;
    }
    {
      int row = b1 * cT + step;
      const float* xp = xproj + (size_t)row * cG + j0;
      float gi = xp[0 * cH + jj1] + gbuf[b1][0 * 16 + jj1];
      float gf = xp[1 * cH + jj1] + gbuf[b1][1 * 16 + jj1];
      float gg = xp[2 * cH + jj1] + gbuf[b1][2 * 16 + jj1];
      float go = xp[3 * cH + jj1] + gbuf[b1][3 * 16 + jj1];
      c1 = sigm(gf) * c1 + sigm(gi) * tanhf(gg);
      float h = sigm(go) * tanhf(c1);
      hn[b1 * cH + j0 + jj1] = h;
      hs[(size_t)row * cH + j0 + jj1] = h;
    }

    __syncthreads();
    __threadfence();                               // release h writes
    if (tid == 0) {
      atomicAdd(ctr, 1u);
      unsigned target = 32u * (unsigned)(step + 1);
      volatile unsigned* vc = ctr;
      while (*vc < target) { __builtin_amdgcn_s_sleep(2); }
    }
    __syncthreads();
    __threadfence();                               // acquire other WGs' h
  }
}

// ---------------------------------------------------------------------------
// K4: fused vocab GEMMs + online logsumexp.  64 (b,t) rows per WG, vocab tiles
// of 64.  8 waves = 4 N-tiles x 2 (M-tile pairs); each wave computes two
// M-tiles sharing one streamed weight fragment.
//   s1  = h  @ w_prior.T           (K=512)
//   lp  = [z|z2] @ Cmat.T + constv (K=512)
//   gen = z  @ w_gen.T             (K=256)
// L1=lse(s1+bp), L2=lse(s1+bp+lp+const), L3=lse(gen+bg); gather gen[x].
// Bias rows (b_prior/constv/b_gen) double-buffered in LDS via TDM prefetch.
// ---------------------------------------------------------------------------
__global__ void __launch_bounds__(256) vocab_kernel(const float* __restrict__ hs,
                                                    const float* __restrict__ zc,
                                                    const float* __restrict__ w_prior,
                                                    const float* __restrict__ b_prior,
                                                    const float* __restrict__ Cmat,
                                                    const float* __restrict__ constv,
                                                    const float* __restrict__ w_gen,
                                                    const float* __restrict__ b_gen,
                                                    const int* __restrict__ xg,
                                                    const int* __restrict__ xsl,
                                                    const float* __restrict__ qz,
                                                    float* __restrict__ net) {
  const int tid = threadIdx.x, lane = tid & 31, w = tid >> 5;
  const int ni = w >> 1;                 // N-tile 0..3
  const int rbase = (w & 1) * 32;        // two M-tiles: rows rbase..+15, +16..+31
  const int nm = lane & 15, koff = (lane >> 4) << 1, mb = (lane >> 4) << 3;
  const int row0 = blockIdx.x * 64;

  __shared__ float sA[64][68], sB[64][68], sC[64][68];   // s1 / lp / gen tiles
  __shared__ float pm[3][64][4], psum[3][64][4];         // per-chunk partials
  __shared__ float rm[3][64], rs[3][64];                 // running (max,sum)
  __shared__ float gxbuf[64];                            // gen logit at target
  __shared__ float biasbuf[2][3][64];                    // TDM double buffer

  if (tid < 192) { rm[tid >> 6][tid & 63] = -INFINITY; rs[tid >> 6][tid & 63] = 0.f; }
  if (tid < 64) gxbuf[tid] = 0.f;

#ifdef USE_TDM
  if (w == 0) {                                          // prologue: tile 0
    tdm_load_row(b_prior, 64, &biasbuf[0][0][0]);
    tdm_load_row(constv,  64, &biasbuf[0][1][0]);
    tdm_load_row(b_gen,   64, &biasbuf[0][2][0]);
  }
#endif
  __syncthreads();

  const float* apH0 = hs + (size_t)(row0 + rbase + nm) * 512 + koff;
  const float* apH1 = apH0 + (size_t)16 * 512;
  const float* apZ0 = zc + (size_t)(row0 + rbase + nm) * 512 + koff;
  const float* apZ1 = apZ0 + (size_t)16 * 512;

  int it = 0;
  for (int v0 = 0; v0 < cV; v0 += 64, ++it) {
    const int cur = it & 1;

#ifdef USE_TDM
    {                                                    // prefetch tile i+1
      int vn = (v0 + 64 < cV) ? v0 + 64 : v0;            // clamp: harmless refetch
      int remn = cV - vn; if (remn > 64) remn = 64;
      if (w == 0) {
        tdm_load_row(b_prior + vn, remn, &biasbuf[cur ^ 1][0][0]);
        tdm_load_row(constv  + vn, remn, &biasbuf[cur ^ 1][1][0]);
        tdm_load_row(b_gen   + vn, remn, &biasbuf[cur ^ 1][2][0]);
      }
    }
#endif

    int vcol = v0 + ni * 16 + nm;
    if (vcol >= cV) vcol = cV - 1;                       // clamp (masked later)
    int vpf = vcol + 64; if (vpf >= cV) vpf = cV - 1;
    __builtin_prefetch(w_prior + (size_t)vpf * 512 + koff, 0, 1);
    __builtin_prefetch(Cmat    + (size_t)vpf * 512 + koff, 0, 1);
    __builtin_prefetch(w_gen   + (size_t)vpf * 256 + koff, 0, 1);

    v8f aS0 = {}, aS1 = {};
    wmma_span2(apH0, apH1, w_prior + (size_t)vcol * 512 + koff, 512, aS0, aS1);
    v8f aL0 = {}, aL1 = {};
    wmma_span2(apZ0, apZ1, Cmat + (size_t)vcol * 512 + koff, 512, aL0, aL1);
    v8f aG0 = {}, aG1 = {};
    wmma_span2(apZ0, apZ1, w_gen + (size_t)vcol * 256 + koff, 256, aG0, aG1);

    const int co = ni * 16 + nm;
#pragma unroll
    for (int rr = 0; rr < 8; ++rr) {
      int r0 = rbase + mb + rr, r1 = rbase + 16 + mb + rr;
      sA[r0][co] = aS0[rr]; sA[r1][co] = aS1[rr];
      sB[r0][co] = aL0[rr]; sB[r1][co] = aL1[rr];
      sC[r0][co] = aG0[rr]; sC[r1][co] = aG1[rr];
    }

#ifdef USE_TDM
    if (w == 0) __builtin_amdgcn_s_wait_tensorcnt(3);    // tile i's 3 done
#else
    if (tid < 192) {                                     // fallback: direct load
      int mat = tid >> 6, c = tid & 63, v = v0 + c;
      float val = 0.f;
      if (v < cV)
        val = (mat == 0) ? b_prior[v] : (mat == 1) ? constv[v] : b_gen[v];
      biasbuf[cur][mat][c] = val;
    }
#endif
    __syncthreads();

    // per-thread online lse over 16 columns of one row, three matrices
    {
      int ro = tid & 63, ch = tid >> 6;
      float m0 = -INFINITY, s0 = 0.f;
      float m1 = -INFINITY, s1 = 0.f;
      float m2 = -INFINITY, s2 = 0.f;
#pragma unroll
      for (int j = 0; j < 16; ++j) {
        int c = ch * 16 + j, v = v0 + c;
        if (v < cV) {
          float vs = sA[ro][c] + biasbuf[cur][0][c];
          float vl = vs + sB[ro][c] + biasbuf[cur][1][c];
          float vg = sC[ro][c] + biasbuf[cur][2][c];
          if (vs > m0) { s0 = s0 * expf(m0 - vs) + 1.f; m0 = vs; } else s0 += expf(vs - m0);
          if (vl > m1) { s1 = s1 * expf(m1 - vl) + 1.f; m1 = vl; } else s1 += expf(vl - m1);
          if (vg > m2) { s2 = s2 * expf(m2 - vg) + 1.f; m2 = vg; } else s2 += expf(vg - m2);
        }
      }
      pm[0][ro][ch] = m0; psum[0][ro][ch] = s0;
      pm[1][ro][ch] = m1; psum[1][ro][ch] = s1;
      pm[2][ro][ch] = m2; psum[2][ro][ch] = s2;
    }
    __syncthreads();

    if (tid < 192) {
      int mat = tid >> 6, ro = tid & 63;
      float M = -INFINITY;
#pragma unroll
      for (int j = 0; j < 4; ++j) M = fmaxf(M, pm[mat][ro][j]);
      if (M > -INFINITY) {
        float S = 0.f;
#pragma unroll
        for (int j = 0; j < 4; ++j) {
          float mj = pm[mat][ro][j];
          if (mj > -INFINITY) S += psum[mat][ro][j] * expf(mj - M);
        }
        float cm = rm[mat][ro], cs = rs[mat][ro];
        if (M > cm) { rs[mat][ro] = cs * expf(cm - M) + S; rm[mat][ro] = M; }
        else        { rs[mat][ro] = cs + S * expf(M - cm); }
      }
      if (mat == 2) {
        int vt = xg[row0 + ro];
        if (vt >= v0 && vt < v0 + 64)
          gxbuf[ro] = sC[ro][vt - v0] + biasbuf[cur][2][vt - v0];
      }
    }
    __syncthreads();
  }

  if (tid < 64) {
    int g = row0 + tid;
    float L1 = rm[0][tid] + logf(rs[0][tid]);
    float L2 = rm[1][tid] + logf(rs[1][tid]);
    float L3 = rm[2][tid] + logf(rs[2][tid]);
    int b = g >> 7, tt = g & 127;
    float mask = (tt < xsl[b]) ? 1.f : 0.f;
    float pz  = L2 - L1;
    float kl  = qz[g] - pz;
    float lpx = gxbuf[tid] - L3;
    net[g] = mask * (lpx - kl);
  }
}

// ---------------------------------------------------------------------------
// K5: final deterministic reduction -> loss scalar
// ---------------------------------------------------------------------------
__global__ void __launch_bounds__(256) loss_kernel(const float* __restrict__ net,
                                                   const int* __restrict__ xsl,
                                                   float* __restrict__ out) {
  int tid = threadIdx.x;
  float s = 0.f;
  for (int i = tid; i < cBT; i += 256) s += net[i];
  __shared__ float red[256];
  red[tid] = s;
  __syncthreads();
  for (int st = 128; st > 0; st >>= 1) {
    if (tid < st) red[tid] += red[tid + st];
    __syncthreads();
  }
  if (tid == 0) {
    int sls = 0;
    for (int i = 0; i < cB; ++i) sls += xsl[i];
    out[0] = -(red[0] / (float)sls);
  }
}

// ---------------------------------------------------------------------------
extern "C" void kernel_launch(void* const* d_in, const int* in_sizes, int n_in,
                              void* d_out, int out_size, void* d_ws, size_t ws_size,
                              hipStream_t stream) {
  (void)in_sizes; (void)n_in; (void)out_size; (void)ws_size;
  const int*   x       = (const int*)d_in[0];
  const int*   xsl     = (const int*)d_in[1];
  const float* eps     = (const float*)d_in[2];
  const float* emb     = (const float*)d_in[3];
  const float* w_ih    = (const float*)d_in[4];
  const float* w_hh    = (const float*)d_in[5];
  const float* b_ih    = (const float*)d_in[6];
  const float* b_hh    = (const float*)d_in[7];
  const float* w_prior = (const float*)d_in[8];
  const float* b_prior = (const float*)d_in[9];
  const float* w_gen   = (const float*)d_in[10];
  const float* b_gen   = (const float*)d_in[11];
  float* out = (float*)d_out;

  char* p = (char*)d_ws;
  auto alloc = [&](size_t bytes) -> void* {
    void* q = (void*)p;
    p += (bytes + 255) & ~size_t(255);
    return q;
  };
  float*    Cmat   = (float*)alloc((size_t)cV * 512 * 4);   // 20.5 MB
  float*    constv = (float*)alloc((size_t)cV * 4);
  float*    zc     = (float*)alloc((size_t)cBT * 512 * 4);  //  8.4 MB
  float*    qz     = (float*)alloc((size_t)cBT * 4);
  float*    xproj  = (float*)alloc((size_t)cBT * cG * 4);   // 33.6 MB
  float*    hb0    = (float*)alloc((size_t)cB * cH * 4);
  float*    hb1    = (float*)alloc((size_t)cB * cH * 4);
  float*    hsbuf  = (float*)alloc((size_t)cBT * cH * 4);   //  8.4 MB
  float*    net    = (float*)alloc((size_t)cBT * 4);
  unsigned* ctr    = (unsigned*)alloc(256);

  prior_kernel<<<cV, 256, 0, stream>>>(emb, Cmat, constv);
  z_kernel<<<cBT, 256, 0, stream>>>(x, xsl, eps, emb, zc, qz);
  init_kernel<<<(cB * cH + 255) / 256, 256, 0, stream>>>(hb0, hb1, ctr);
  xproj_kernel<<<dim3(cG / 64, cBT / 32), 256, 0, stream>>>(zc, w_ih, b_ih, b_hh, xproj);
  lstm_kernel<<<32, 256, 0, stream>>>(xproj, w_hh, hb0, hb1, hsbuf, ctr);
  vocab_kernel<<<cBT / 64, 256, 0, stream>>>(hsbuf, zc, w_prior, b_prior, Cmat,
                                             constv, w_gen, b_gen, x, xsl, qz, net);
  loss_kernel<<<1, 256, 0, stream>>>(net, xsl, out);
}